// SvmLoss_11261404250863
// MI455X (gfx1250) — compile-verified
//
#include <hip/hip_runtime.h>
#include <hip/hip_bf16.h>

// ---------------------------------------------------------------------------
// SVM rank loss (pairwise squared hinge, masked) + regression term, N=8192.
// Strategy: rank-2 factorization H[i,j] = 1 + lg_i - lg_j = A(16x4) x B(4x16)
// computed with v_wmma_f32_16x16x4_f32 (one WMMA = 256 pair pre-activations),
// then VALU applies mask / relu / square / accumulate. j-side data staged in
// LDS chunks; deterministic tree reductions (no atomics).
// ---------------------------------------------------------------------------

typedef __attribute__((ext_vector_type(2))) float v2f;
typedef __attribute__((ext_vector_type(8))) float v8f;

#define RANK_RATIO_F 1.0f
#define THREADS 256
#define WAVES_PER_BLOCK 8   // 256 / wave32
#define CHUNK 2048          // j-chunk staged in LDS (2 * 8 KB)

#if __has_builtin(__builtin_amdgcn_wmma_f32_16x16x4_f32)
#define HAVE_WMMA_F32X4 1
#endif

__global__ __launch_bounds__(THREADS) void svm_pair_kernel(
    const float* __restrict__ logit,
    const float* __restrict__ status_time,
    float* __restrict__ blkSum,
    int*   __restrict__ blkCnt,
    int N)
{
    __shared__ float lgJ[CHUNK];   // -logit[j]
    __shared__ float tJ[CHUNK];    // time[j]
    __shared__ float redS[THREADS];
    __shared__ int   redC[THREADS];

    const int tid   = threadIdx.x;
    const int lane  = tid & 31;
    const int wave  = tid >> 5;
    const int iTile = blockIdx.x * WAVES_PER_BLOCK + wave; // 16-row i tile
    const int i0    = iTile * 16;
    const bool lo   = lane < 16;

    // --- A matrix (16x4 f32): A[m,0]=lg_i[m], A[m,1]=1, A[m,2..3]=0 ---
    // Layout (ISA 7.12.2): VGPR0 = K0 (lanes 0-15) / K2 (lanes 16-31),
    //                      VGPR1 = K1 (lanes 0-15) / K3 (lanes 16-31).
    v2f A;
    A.x = lo ? -logit[i0 + lane] : 0.0f;  // K0 = lg_i ; K2 = 0
    A.y = lo ? 1.0f : 0.0f;               // K1 = 1    ; K3 = 0

    // Per-lane i-row data for mask: element (r, lane) is row m = r + 8*(lane>=16)
    const int mbase = i0 + ((lane >= 16) ? 8 : 0);
    float ti[8], si[8];
#ifndef HAVE_WMMA_F32X4
    float lgi8[8];
#endif
#pragma unroll
    for (int r = 0; r < 8; ++r) {
        float2 st = ((const float2*)status_time)[mbase + r];
        si[r] = st.x;
        ti[r] = st.y;
#ifndef HAVE_WMMA_F32X4
        lgi8[r] = -logit[mbase + r];
#endif
    }

    float acc = 0.0f;
    int   cnt = 0;

    for (int base = 0; base < N; base += CHUNK) {
        __syncthreads();
        for (int k = tid; k < CHUNK; k += THREADS) {
            int idx = base + k;
            bool ok = idx < N;
            lgJ[k] = ok ? -logit[idx] : 0.0f;
            tJ[k]  = ok ? status_time[2 * idx + 1] : -3.4e38f; // mask always false OOB
        }
        __syncthreads();

        for (int j0 = 0; j0 < CHUNK; j0 += 16) {
            const int   n   = lane & 15;         // this lane's column in the tile
            const float lgj = lgJ[j0 + n];
            const float tj  = tJ[j0 + n];

#ifdef HAVE_WMMA_F32X4
            // --- B matrix (4x16 f32): B[0,n]=1, B[1,n]=1-lg_j[n], B[2..3,n]=0 ---
            v2f B;
            B.x = lo ? 1.0f : 0.0f;              // K0 = 1       ; K2 = 0
            B.y = lo ? (1.0f - lgj) : 0.0f;      // K1 = 1-lg_j  ; K3 = 0
            v8f C = {};
            // D[m,n] = lg_i[m] + (1 - lg_j[n])  (full 16x16 pre-hinge tile)
            v8f D = __builtin_amdgcn_wmma_f32_16x16x4_f32(
                false, A, false, B, (short)0, C, false, false);
#endif
#pragma unroll
            for (int r = 0; r < 8; ++r) {
#ifdef HAVE_WMMA_F32X4
                float h = D[r];
#else
                float h = 1.0f + lgi8[r] - lgj;
#endif
                float hp = fmaxf(h, 0.0f);
                bool  m  = (si[r] == 1.0f) & (ti[r] < tj);
                float v  = m ? hp : 0.0f;
                acc = fmaf(v, v, acc);
                cnt += m ? 1 : 0;
            }
        }
    }

    // Deterministic block tree reduction
    redS[tid] = acc;
    redC[tid] = cnt;
    __syncthreads();
    for (int s = THREADS / 2; s > 0; s >>= 1) {
        if (tid < s) {
            redS[tid] += redS[tid + s];
            redC[tid] += redC[tid + s];
        }
        __syncthreads();
    }
    if (tid == 0) {
        blkSum[blockIdx.x] = redS[0];
        blkCnt[blockIdx.x] = redC[0];
    }
}

__global__ __launch_bounds__(THREADS) void svm_finalize_kernel(
    const float* __restrict__ logit,
    const float* __restrict__ status_time,
    const float* __restrict__ blkSum,
    const int*   __restrict__ blkCnt,
    float* __restrict__ out,
    int nBlocks, int N)
{
    __shared__ float redR[THREADS];
    const int tid = threadIdx.x;

    // Regression term (weight 1-RANK_RATIO == 0, computed for fidelity)
    float racc = 0.0f;
    for (int i = tid; i < N; i += THREADS) {
        float  lg = -logit[i];
        float2 st = ((const float2*)status_time)[i];
        float  u  = st.y - lg;
        float  cp = fmaxf(u, 0.0f);
        float  t  = (1.0f - st.x) * cp + st.x * u;
        racc = fmaf(t, t, racc);
    }
    redR[tid] = racc;
    __syncthreads();
    for (int s = THREADS / 2; s > 0; s >>= 1) {
        if (tid < s) redR[tid] += redR[tid + s];
        __syncthreads();
    }

    if (tid == 0) {
        double    ssum = 0.0;
        long long c    = 0;
        for (int b = 0; b < nBlocks; ++b) {
            ssum += (double)blkSum[b];
            c    += (long long)blkCnt[b];
        }
        if (c < 1) c = 1;
        float rank = (float)(ssum / (double)c);
        float reg  = redR[0] / (float)N;
        out[0] = RANK_RATIO_F * rank + (1.0f - RANK_RATIO_F) * reg;
    }
}

extern "C" void kernel_launch(void* const* d_in, const int* in_sizes, int n_in,
                              void* d_out, int out_size, void* d_ws, size_t ws_size,
                              hipStream_t stream)
{
    const float* logit       = (const float*)d_in[0];     // (N,1) f32
    const float* status_time = (const float*)d_in[1];     // (N,2) f32
    const int    N           = in_sizes[0];               // 8192

    const int iTiles  = N / 16;                           // 512
    int       nBlocks = iTiles / WAVES_PER_BLOCK;         // 64
    if (nBlocks < 1) nBlocks = 1;

    float* blkSum = (float*)d_ws;
    int*   blkCnt = (int*)(blkSum + nBlocks);

    svm_pair_kernel<<<nBlocks, THREADS, 0, stream>>>(
        logit, status_time, blkSum, blkCnt, N);
    svm_finalize_kernel<<<1, THREADS, 0, stream>>>(
        logit, status_time, blkSum, blkCnt, (float*)d_out, nBlocks, N);
}